// DeformConv2d_58652073394264
// MI455X (gfx1250) — compile-verified
//
#include <hip/hip_runtime.h>

typedef __attribute__((ext_vector_type(2))) float v2f;
typedef __attribute__((ext_vector_type(8))) float v8f;

// Problem constants (from reference)
constexpr int Bn   = 4;
constexpr int Cn   = 64;     // input channels
constexpr int On   = 64;     // output channels
constexpr int Hn   = 128;
constexpr int Wn   = 128;
constexpr int HWn  = Hn * Wn;        // 16384
constexpr int Gn   = 8;              // offset groups
constexpr int CGn  = Cn / Gn;        // 8 channels per group
constexpr int Kt   = 9;              // 3x3 taps
constexpr int QTOT = Cn * Kt;        // 576 = GEMM reduction dim
constexpr int NPIX = 32;             // pixels per workgroup (one row segment)

__global__ __launch_bounds__(256) void dcnv2_fused_wmma(
    const float* __restrict__ input,   // [B,64,128,128]
    const float* __restrict__ offset,  // [B,144,128,128]
    const float* __restrict__ mask,    // [B,72,128,128]
    const float* __restrict__ weight,  // [64,64,3,3] == [O][q=c*9+k]
    const float* __restrict__ bias,    // [64]
    float* __restrict__ out)           // [B,64,128,128]
{
    extern __shared__ float col[];     // [QTOT][NPIX] = 576*32 floats = 72KB

    const int b   = blockIdx.y;
    const int p0  = blockIdx.x * NPIX; // tile of 32 consecutive pixels (within one row)
    const int row = p0 >> 7;           // ho (W=128)
    const int wo0 = p0 & (Wn - 1);
    const int tid = threadIdx.x;

    const float* inB  = input  + (size_t)b * Cn * HWn;
    const float* offB = offset + (size_t)b * (2 * Gn * Kt) * HWn;
    const float* mskB = mask   + (size_t)b * (Gn * Kt) * HWn;

    // ---------------- Phase 1: bilinear im2col into LDS ----------------
    // jobs = (g,k) x pixel : 72 * 32 = 2304 positions, 9 per thread
    for (int job = tid; job < Gn * Kt * NPIX; job += 256) {
        const int n  = job & (NPIX - 1);
        const int gk = job >> 5;                  // g*9 + k
        const int g  = gk / Kt;
        const int k  = gk - g * Kt;
        const int ky = k / 3;
        const int kx = k - ky * 3;
        const int ho = row;
        const int wo = wo0 + n;
        const int pix = ho * Wn + wo;

        const float dy = offB[(size_t)(gk * 2    ) * HWn + pix];
        const float dx = offB[(size_t)(gk * 2 + 1) * HWn + pix];
        const float mm = mskB[(size_t)gk * HWn + pix];

        const float py = (float)(ho - 1 + ky) + dy;   // pad=1, dil=1, stride=1
        const float px = (float)(wo - 1 + kx) + dx;
        const float fy0 = floorf(py);
        const float fx0 = floorf(px);
        const float ly = py - fy0;
        const float lx = px - fx0;
        const int y0 = (int)fy0, x0 = (int)fx0;
        const int y1 = y0 + 1,   x1 = x0 + 1;

        const bool vy0 = (y0 >= 0) && (y0 < Hn);
        const bool vy1 = (y1 >= 0) && (y1 < Hn);
        const bool vx0 = (x0 >= 0) && (x0 < Wn);
        const bool vx1 = (x1 >= 0) && (x1 < Wn);
        const int cy0 = min(max(y0, 0), Hn - 1);
        const int cy1 = min(max(y1, 0), Hn - 1);
        const int cx0 = min(max(x0, 0), Wn - 1);
        const int cx1 = min(max(x1, 0), Wn - 1);

        const float w00 = (1.f - ly) * (1.f - lx) * ((vy0 && vx0) ? 1.f : 0.f);
        const float w01 = (1.f - ly) * lx         * ((vy0 && vx1) ? 1.f : 0.f);
        const float w10 = ly * (1.f - lx)         * ((vy1 && vx0) ? 1.f : 0.f);
        const float w11 = ly * lx                 * ((vy1 && vx1) ? 1.f : 0.f);

        const int i00 = cy0 * Wn + cx0;
        const int i01 = cy0 * Wn + cx1;
        const int i10 = cy1 * Wn + cx0;
        const int i11 = cy1 * Wn + cx1;

        const float* chp = inB + (size_t)(g * CGn) * HWn;
        #pragma unroll
        for (int c = 0; c < CGn; ++c) {
            const float* pl = chp + (size_t)c * HWn;
            const float s = w00 * pl[i00] + w01 * pl[i01]
                          + w10 * pl[i10] + w11 * pl[i11];
            col[((g * CGn + c) * Kt + k) * NPIX + n] = mm * s;
        }
    }
    __syncthreads();

    // ---------------- Phase 2: WMMA GEMM (fp32, 16x16x4) ----------------
    // 8 waves: 4 M-tiles (16 out channels) x 2 N-tiles (16 pixels)
    const int wave  = tid >> 5;
    const int lane  = tid & 31;
    const int mtile = wave & 3;
    const int ntile = wave >> 2;
    const int lo16  = lane & 15;
    const int kb    = (lane >> 4) * 2;  // lanes 0-15 -> K {0,1}; 16-31 -> K {2,3}
    const int m     = mtile * 16 + lo16;
    const int ncol  = ntile * 16 + lo16;

    v8f acc = {};
    const float* wrow = weight + (size_t)m * QTOT;
    #pragma unroll 4
    for (int q = 0; q < QTOT; q += 4) {
        // A fragment: 16x4 slab of weight, aligned float2 per lane
        const v2f a = *(const v2f*)(wrow + q + kb);
        // B fragment: 4x16 slab of the LDS column tile
        v2f bb;
        bb.x = col[(q + kb    ) * NPIX + ncol];
        bb.y = col[(q + kb + 1) * NPIX + ncol];
        acc = __builtin_amdgcn_wmma_f32_16x16x4_f32(
            /*neg_a=*/false, a, /*neg_b=*/false, bb,
            /*c_mod=*/(short)0, acc, /*reuse_a=*/false, /*reuse_b=*/false);
    }

    // ---------------- Epilogue: C/D layout -> NCHW + bias ----------------
    const int p     = p0 + ntile * 16 + lo16;       // pixel this lane owns
    const int mhalf = (lane >> 4) * 8;              // lanes 16-31 hold M=r+8
    #pragma unroll
    for (int r = 0; r < 8; ++r) {
        const int mo = mtile * 16 + mhalf + r;
        out[((size_t)b * On + mo) * HWn + p] = acc[r] + bias[mo];
    }
}

extern "C" void kernel_launch(void* const* d_in, const int* in_sizes, int n_in,
                              void* d_out, int out_size, void* d_ws, size_t ws_size,
                              hipStream_t stream) {
    const float* input  = (const float*)d_in[0];
    const float* offset = (const float*)d_in[1];
    const float* mask   = (const float*)d_in[2];
    const float* weight = (const float*)d_in[3];
    const float* bias   = (const float*)d_in[4];
    float* out = (float*)d_out;

    dim3 grid(HWn / NPIX, Bn);                       // 512 x 4 workgroups
    const size_t lds_bytes = (size_t)QTOT * NPIX * sizeof(float);  // 73728 B
    dcnv2_fused_wmma<<<grid, 256, lds_bytes, stream>>>(
        input, offset, mask, weight, bias, out);
}